// self_LSTM_sparse_attn_predict_eval_interpre_45251775430861
// MI455X (gfx1250) — compile-verified
//
#include <hip/hip_runtime.h>
#include <hip/hip_bf16.h>
#include <math.h>

typedef __attribute__((ext_vector_type(2))) float v2f;
typedef __attribute__((ext_vector_type(8))) float v8f;

#define B_   64
#define T_   256
#define I_   64
#define H_   128
#define TOPK 10
#define EPSF 1e-7f

// ---------------------------------------------------------------------------
// Kernel 1: Ht[m,h] = tanh( X[m,:] @ W[:,h] ),  m in [0, B*T)  (16384 x 64 x 128)
// fp32 WMMA 16x16x4, one 16x16 output tile per wave32.
//   A (16x4 f32, 2 VGPR): lanes 0-15 hold M=0..15 {K=k+0,k+1}; lanes 16-31 {K=k+2,k+3}
//   B (4x16 f32, 2 VGPR): lanes 0-15 rows K=k+0,k+1 (N=lane); lanes 16-31 rows K=k+2,k+3
//   D (16x16 f32, 8 VGPR): VGPR v: lanes 0-15 -> M=v, lanes 16-31 -> M=v+8, N=lane&15
// ---------------------------------------------------------------------------
__global__ __launch_bounds__(128)
void k_gemm_tanh(const float* __restrict__ X, const float* __restrict__ W,
                 float* __restrict__ Ht)
{
    const int lane = threadIdx.x & 31;
    const int wid  = threadIdx.x >> 5;
    const int tile = blockIdx.x * 4 + wid;      // (B*T/16)*(H/16) = 8192 tiles
    const int NT   = H_ / 16;                   // 8 tiles along N
    const int m0   = (tile / NT) * 16;
    const int n0   = (tile % NT) * 16;
    const int r    = lane & 15;                 // M index (for A) / N index (for B,D)
    const int hi   = lane >> 4;                 // half-wave selector

    v8f c = {};
    #pragma unroll
    for (int k = 0; k < I_; k += 4) {
        const int ka = k + hi * 2;
        v2f a, b;
        a.x = X[(m0 + r) * I_ + ka + 0];
        a.y = X[(m0 + r) * I_ + ka + 1];
        b.x = W[(ka + 0) * H_ + n0 + r];
        b.y = W[(ka + 1) * H_ + n0 + r];
        c = __builtin_amdgcn_wmma_f32_16x16x4_f32(false, a, false, b,
                                                  (short)0, c, false, false);
    }
    #pragma unroll
    for (int v = 0; v < 8; ++v) {
        const int m = m0 + v + hi * 8;
        Ht[m * H_ + n0 + r] = tanhf(c[v]);
    }
}

// ---------------------------------------------------------------------------
// Kernel 2: Abase[m] = sum_h tanh(Ht[m,h]) * w_t[h]   (the per-step additive
// constant of the attention score; only first half of w_t).
// One wave32 per row, shfl_xor reduction.
// ---------------------------------------------------------------------------
__global__ __launch_bounds__(128)
void k_attn_base(const float* __restrict__ Ht, const float* __restrict__ w_t,
                 float* __restrict__ Abase)
{
    const int lane = threadIdx.x & 31;
    const int wid  = threadIdx.x >> 5;
    const int row  = blockIdx.x * 4 + wid;      // [0, B*T)
    float s = 0.f;
    #pragma unroll
    for (int j = 0; j < H_; j += 32) {
        const int h = j + lane;
        s += tanhf(Ht[row * H_ + h]) * w_t[h];
    }
    #pragma unroll
    for (int m = 16; m > 0; m >>= 1) s += __shfl_xor(s, m, 32);
    if (lane == 0) Abase[row] = s;
}

// ---------------------------------------------------------------------------
// Kernel 3: the serial recurrence. One workgroup (128 threads = 4 waves) per
// batch; h_buf (257x128 f32 = 131.5 KB) lives entirely in CDNA5's 320KB LDS.
// Key identity: attn[t] = Abase[b,i] + c[t] with c[t] fixed once row t exists,
// and Abase cancels in the top-k threshold weights -> maintain a running
// sorted top-10 of c incrementally; each step touches <=10 h_buf rows.
// ---------------------------------------------------------------------------
__global__ __launch_bounds__(128)
void k_recurrence(const float* __restrict__ Ht, const float* __restrict__ Abase,
                  const float* __restrict__ w_t, float* __restrict__ out)
{
    __shared__ float hbuf[(T_ + 1) * H_];   // 131584 B
    __shared__ float cbuf[T_ + 1];
    __shared__ float topv[TOPK];
    __shared__ int   topi[TOPK];
    __shared__ float selw[TOPK];
    __shared__ int   seli[TOPK];
    __shared__ int   nsel;
    __shared__ float partial[4];

    const int b    = blockIdx.x;
    const int tid  = threadIdx.x;           // == h index, 0..127
    const int lane = tid & 31;
    const int wid  = tid >> 5;

    // row 0 of h_buf is the only row read before being written -> zero it
    hbuf[tid] = 0.f;
    if (tid == 0) {
        cbuf[0] = 0.f;
        topv[0] = 0.f; topi[0] = 0;         // c[0] = 0 enters the top-10 list
        for (int j = 1; j < TOPK; ++j) { topv[j] = -3.0e38f; topi[j] = -1; }
    }
    const float wt2 = w_t[H_ + tid];        // second half of w_t
    __syncthreads();

    const float* HtRow = Ht + (size_t)b * T_ * H_;

    for (int i = 0; i < T_; ++i) {
        const float hti = HtRow[i * H_ + tid];
        if (i + 1 < T_) __builtin_prefetch(&HtRow[(i + 1) * H_ + tid], 0, 1);

        if (tid == 0) {
            int n;
            if (i < TOPK) {
                // raw path (i+1 <= TOP_K): weights are the raw scores A + c[t]
                const float Ab = Abase[b * T_ + i];
                n = i + 1;
                for (int t = 0; t <= i; ++t) { seli[t] = t; selw[t] = Ab + cbuf[t]; }
            } else {
                // sparse path: aw[t] = max(c[t] - c10 - eps, 0), normalized.
                // Only the top-9 c values can be positive.
                const float c10 = topv[TOPK - 1];
                float ssum = 0.f; n = 0;
                for (int j = 0; j < TOPK - 1; ++j) {
                    const float wv = topv[j] - c10 - EPSF;
                    if (wv > 0.f) { selw[n] = wv; seli[n] = topi[j]; ssum += wv; }
                }
                const float inv = 1.0f / (ssum + EPSF);
                for (int j = 0; j < n; ++j) selw[j] *= inv;
            }
            nsel = n;
        }
        __syncthreads();

        float acc = 0.f;
        const int n = nsel;
        for (int j = 0; j < n; ++j) acc += selw[j] * hbuf[seli[j] * H_ + tid];

        if (i == T_ - 1) out[b * H_ + tid] = acc;   // scan returns last attn_c

        const float hn = hti + acc;                 // h_new = h_t + attn_c
        hbuf[(i + 1) * H_ + tid] = hn;

        // c[i+1] = sum_h tanh(h_new[h]) * w_t[H+h]
        float p = tanhf(hn) * wt2;
        #pragma unroll
        for (int m = 16; m > 0; m >>= 1) p += __shfl_xor(p, m, 32);
        if (lane == 0) partial[wid] = p;
        __syncthreads();

        if (tid == 0) {
            const float cn = partial[0] + partial[1] + partial[2] + partial[3];
            cbuf[i + 1] = cn;
            // insert into sorted (desc) running top-10 of c
            float v = cn; int idx = i + 1;
            for (int j = 0; j < TOPK; ++j) {
                if (v > topv[j]) {
                    const float tv = topv[j]; topv[j] = v; v = tv;
                    const int   ti = topi[j]; topi[j] = idx; idx = ti;
                }
            }
        }
        __syncthreads();
    }
}

// ---------------------------------------------------------------------------
extern "C" void kernel_launch(void* const* d_in, const int* in_sizes, int n_in,
                              void* d_out, int out_size, void* d_ws, size_t ws_size,
                              hipStream_t stream)
{
    const float* x  = (const float*)d_in[0];   // (B, T, I) fp32
    const float* w  = (const float*)d_in[1];   // (I, H)    fp32
    const float* wt = (const float*)d_in[2];   // (2H, 1)   fp32
    float* out = (float*)d_out;                // (B, H)    fp32

    float* Ht = (float*)d_ws;                          // B*T*H floats (8.4 MB)
    float* Ab = Ht + (size_t)B_ * T_ * H_;             // B*T floats

    // 1) Ht = tanh(X @ W) via fp32 WMMA: 8192 tiles, 4 waves/block
    k_gemm_tanh<<<(B_ * T_ / 16) * (H_ / 16) / 4, 128, 0, stream>>>(x, w, Ht);
    // 2) per-row score base
    k_attn_base<<<(B_ * T_) / 4, 128, 0, stream>>>(Ht, wt, Ab);
    // 3) serial sparse-attention recurrence, one WGP per batch
    k_recurrence<<<B_, 128, 0, stream>>>(Ht, Ab, wt, out);
}